// GATModel_44470091383467
// MI455X (gfx1250) — compile-verified
//
#include <hip/hip_runtime.h>
#include <hip/hip_bf16.h>
#include <math.h>

typedef __attribute__((ext_vector_type(2))) float v2f;
typedef __attribute__((ext_vector_type(8))) float v8f;

#define NNODES 100000
#define NFEAT  128
#define NHID   32
#define NCLASS 40
#define NEDGES 1600000
#define NEG_SLOPE 0.2f

static inline int ceil_div(long long a, long long b) { return (int)((a + b - 1) / b); }

// ---------------------------------------------------------------------------
// GEMM1: h1[N,32] = x[N,128] @ W1[128,32]   (fp32 WMMA 16x16x4)
// One wave -> 16 rows, 2 N-tiles. Block = 8 waves = 128 rows.
// ---------------------------------------------------------------------------
__global__ __launch_bounds__(256) void gemm1_wmma(const float* __restrict__ x,
                                                  const float* __restrict__ W,
                                                  float* __restrict__ h) {
    __shared__ float ldsW[NFEAT * NHID];  // 16 KB
    for (int i = threadIdx.x; i < NFEAT * NHID; i += 256) ldsW[i] = W[i];
    __syncthreads();

    const int lane    = threadIdx.x & 31;
    const int wave    = threadIdx.x >> 5;
    const int rowBase = blockIdx.x * 128 + wave * 16;
    const int half    = lane >> 4;   // K-half selector
    const int l16     = lane & 15;   // M for A-frag, N for B/C-frag

    int row  = rowBase + l16;
    int rowc = row < NNODES ? row : (NNODES - 1);
    const float* xr = x + (long long)rowc * NFEAT;

    v8f c0 = {}; v8f c1 = {};
    for (int k = 0; k < NFEAT; k += 4) {
        const int k0 = k + 2 * half;
        v2f a;
        a.x = xr[k0];
        a.y = xr[k0 + 1];
        v2f b0, b1;
        b0.x = ldsW[k0 * NHID + l16];
        b0.y = ldsW[(k0 + 1) * NHID + l16];
        b1.x = ldsW[k0 * NHID + 16 + l16];
        b1.y = ldsW[(k0 + 1) * NHID + 16 + l16];
        c0 = __builtin_amdgcn_wmma_f32_16x16x4_f32(false, a, false, b0, (short)0, c0, false, false);
        c1 = __builtin_amdgcn_wmma_f32_16x16x4_f32(false, a, false, b1, (short)0, c1, false, false);
    }
    #pragma unroll
    for (int m = 0; m < 8; ++m) {
        int r = rowBase + half * 8 + m;
        if (r < NNODES) {
            h[(long long)r * NHID + l16]      = c0[m];
            h[(long long)r * NHID + 16 + l16] = c1[m];
        }
    }
}

// ---------------------------------------------------------------------------
// GEMM2: h2[N,40] = hin[N,32] @ W2[32,40]   (W2 zero-padded to 32x48 in LDS)
// ---------------------------------------------------------------------------
__global__ __launch_bounds__(256) void gemm2_wmma(const float* __restrict__ hin,
                                                  const float* __restrict__ W,
                                                  float* __restrict__ hout) {
    __shared__ float ldsW[NHID * 48];  // 6 KB, cols 40..47 = 0
    for (int i = threadIdx.x; i < NHID * 48; i += 256) {
        int k = i / 48, n = i % 48;
        ldsW[i] = (n < NCLASS) ? W[k * NCLASS + n] : 0.0f;
    }
    __syncthreads();

    const int lane    = threadIdx.x & 31;
    const int wave    = threadIdx.x >> 5;
    const int rowBase = blockIdx.x * 128 + wave * 16;
    const int half    = lane >> 4;
    const int l16     = lane & 15;

    int row  = rowBase + l16;
    int rowc = row < NNODES ? row : (NNODES - 1);
    const float* hr = hin + (long long)rowc * NHID;

    v8f c0 = {}; v8f c1 = {}; v8f c2 = {};
    for (int k = 0; k < NHID; k += 4) {
        const int k0 = k + 2 * half;
        v2f a;
        a.x = hr[k0];
        a.y = hr[k0 + 1];
        v2f b0, b1, b2;
        b0.x = ldsW[k0 * 48 + l16];        b0.y = ldsW[(k0 + 1) * 48 + l16];
        b1.x = ldsW[k0 * 48 + 16 + l16];   b1.y = ldsW[(k0 + 1) * 48 + 16 + l16];
        b2.x = ldsW[k0 * 48 + 32 + l16];   b2.y = ldsW[(k0 + 1) * 48 + 32 + l16];
        c0 = __builtin_amdgcn_wmma_f32_16x16x4_f32(false, a, false, b0, (short)0, c0, false, false);
        c1 = __builtin_amdgcn_wmma_f32_16x16x4_f32(false, a, false, b1, (short)0, c1, false, false);
        c2 = __builtin_amdgcn_wmma_f32_16x16x4_f32(false, a, false, b2, (short)0, c2, false, false);
    }
    #pragma unroll
    for (int m = 0; m < 8; ++m) {
        int r = rowBase + half * 8 + m;
        if (r < NNODES) {
            float* o = hout + (long long)r * NCLASS;
            o[l16]      = c0[m];
            o[16 + l16] = c1[m];
            if (32 + l16 < NCLASS) o[32 + l16] = c2[m];
        }
    }
}

// ---------------------------------------------------------------------------
// Per-node attention logits: as[i] = h[i]·a_src, ad[i] = h[i]·a_dst
// ---------------------------------------------------------------------------
__global__ void alpha_kernel(const float* __restrict__ h,
                             const float* __restrict__ asrc,
                             const float* __restrict__ adst,
                             float* __restrict__ outs, float* __restrict__ outd,
                             int n, int C) {
    int i = blockIdx.x * blockDim.x + threadIdx.x;
    if (i >= n) return;
    const float* r = h + (long long)i * C;
    float s = 0.f, d = 0.f;
    for (int c = 0; c < C; ++c) { float v = r[c]; s += v * asrc[c]; d += v * adst[c]; }
    outs[i] = s;
    outd[i] = d;
}

// ---------------------------------------------------------------------------
// Edge pass: ee = exp(leaky_relu(as[src]+ad[dst])); denom[dst] += ee
// (segment-max elided: softmax is shift-invariant; logits are O(10) here)
// ---------------------------------------------------------------------------
__global__ void edge_kernel(const float* __restrict__ as, const float* __restrict__ ad,
                            const int* __restrict__ src, const int* __restrict__ dst,
                            float* __restrict__ ew, float* __restrict__ denom, int E) {
    int i = blockIdx.x * blockDim.x + threadIdx.x;
    if (i >= E) return;
    float e = as[src[i]] + ad[dst[i]];
    e = e > 0.f ? e : NEG_SLOPE * e;
    float ee = expf(e);
    ew[i] = ee;
    atomicAdd(&denom[dst[i]], ee);
}

// ---------------------------------------------------------------------------
// Scatter: agg[dst, c] += h[src, c] * ew[e]/(denom[dst]+1e-16)
// One thread per (edge, channel).
// ---------------------------------------------------------------------------
__global__ void scatter_kernel(const float* __restrict__ h, const float* __restrict__ ew,
                               const float* __restrict__ denom,
                               const int* __restrict__ src, const int* __restrict__ dst,
                               float* __restrict__ agg, long long total, int C) {
    long long i = (long long)blockIdx.x * blockDim.x + threadIdx.x;
    if (i >= total) return;
    int e = (int)(i / C);
    int c = (int)(i % C);
    int s = src[e], d = dst[e];
    float alpha = ew[e] / (denom[d] + 1e-16f);
    atomicAdd(&agg[(long long)d * C + c], h[(long long)s * C + c] * alpha);
}

// ---------------------------------------------------------------------------
// a[i] = relu(a[i] + b[i % C])
// ---------------------------------------------------------------------------
__global__ void bias_relu_kernel(float* __restrict__ a, const float* __restrict__ b,
                                 long long total, int C) {
    long long i = (long long)blockIdx.x * blockDim.x + threadIdx.x;
    if (i >= total) return;
    float v = a[i] + b[(int)(i % C)];
    a[i] = v > 0.f ? v : 0.f;
}

// ---------------------------------------------------------------------------
// In-place: row = log_softmax(row + b2)
// ---------------------------------------------------------------------------
__global__ void logsoftmax_bias_kernel(float* __restrict__ out,
                                       const float* __restrict__ b, int n) {
    int i = blockIdx.x * blockDim.x + threadIdx.x;
    if (i >= n) return;
    float* row = out + (long long)i * NCLASS;
    float m = -INFINITY;
    for (int c = 0; c < NCLASS; ++c) { float v = row[c] + b[c]; m = fmaxf(m, v); }
    float s = 0.f;
    for (int c = 0; c < NCLASS; ++c) s += expf(row[c] + b[c] - m);
    float lse = m + logf(s);
    for (int c = 0; c < NCLASS; ++c) row[c] = row[c] + b[c] - lse;
}

// ---------------------------------------------------------------------------
extern "C" void kernel_launch(void* const* d_in, const int* in_sizes, int n_in,
                              void* d_out, int out_size, void* d_ws, size_t ws_size,
                              hipStream_t stream) {
    const float* x     = (const float*)d_in[0];
    const int*   src   = (const int*)d_in[1];          // edge_index[0]
    const int*   dst   = src + NEDGES;                 // edge_index[1]
    const float* W1    = (const float*)d_in[2];
    const float* as1v  = (const float*)d_in[3];
    const float* ad1v  = (const float*)d_in[4];
    const float* b1    = (const float*)d_in[5];
    const float* W2    = (const float*)d_in[6];
    const float* as2v  = (const float*)d_in[7];
    const float* ad2v  = (const float*)d_in[8];
    const float* b2    = (const float*)d_in[9];
    float* out = (float*)d_out;

    // Workspace layout (floats): ~49.2 MB total
    float* ws    = (float*)d_ws;
    float* h1    = ws;                                  // NNODES*NHID
    float* as    = h1 + (long long)NNODES * NHID;       // NNODES
    float* ad    = as + NNODES;                         // NNODES
    float* denom = ad + NNODES;                         // NNODES
    float* ew    = denom + NNODES;                      // NEDGES
    float* agg1  = ew + NEDGES;                         // NNODES*NHID
    float* h2    = agg1 + (long long)NNODES * NHID;     // NNODES*NCLASS

    const int T = 256;
    const int gemmBlocks = ceil_div(NNODES, 128);

    // ---- Layer 1 ----
    gemm1_wmma<<<gemmBlocks, T, 0, stream>>>(x, W1, h1);
    alpha_kernel<<<ceil_div(NNODES, T), T, 0, stream>>>(h1, as1v, ad1v, as, ad, NNODES, NHID);
    hipMemsetAsync(denom, 0, sizeof(float) * NNODES, stream);
    hipMemsetAsync(agg1, 0, sizeof(float) * (size_t)NNODES * NHID, stream);
    edge_kernel<<<ceil_div(NEDGES, T), T, 0, stream>>>(as, ad, src, dst, ew, denom, NEDGES);
    {
        long long total = (long long)NEDGES * NHID;
        scatter_kernel<<<ceil_div(total, T), T, 0, stream>>>(h1, ew, denom, src, dst, agg1, total, NHID);
    }
    {
        long long total = (long long)NNODES * NHID;
        bias_relu_kernel<<<ceil_div(total, T), T, 0, stream>>>(agg1, b1, total, NHID);
    }

    // ---- Layer 2 ----
    gemm2_wmma<<<gemmBlocks, T, 0, stream>>>(agg1, W2, h2);
    alpha_kernel<<<ceil_div(NNODES, T), T, 0, stream>>>(h2, as2v, ad2v, as, ad, NNODES, NCLASS);
    hipMemsetAsync(denom, 0, sizeof(float) * NNODES, stream);
    hipMemsetAsync(out, 0, sizeof(float) * (size_t)NNODES * NCLASS, stream);
    edge_kernel<<<ceil_div(NEDGES, T), T, 0, stream>>>(as, ad, src, dst, ew, denom, NEDGES);
    {
        long long total = (long long)NEDGES * NCLASS;
        scatter_kernel<<<ceil_div(total, T), T, 0, stream>>>(h2, ew, denom, src, dst, out, total, NCLASS);
    }
    logsoftmax_bias_kernel<<<ceil_div(NNODES, T), T, 0, stream>>>(out, b2, NNODES);
}